// TH_38783554683567
// MI455X (gfx1250) — compile-verified
//
#include <hip/hip_runtime.h>

// out = x + x * sigmoid(2x), elementwise over 128*64*64*64 floats (2^25).
// HBM-bound streaming kernel: 256 MiB total traffic @ 23.3 TB/s -> ~11.5us floor.
// Hot path: unguarded, 4 x b128 NT loads w/ immediate offsets, exp+rcp+fma, 4 x b128 NT stores.

typedef float v4f __attribute__((ext_vector_type(4)));

__device__ __forceinline__ float gate_elem(float x) {
    // out = x + x * sigmoid(2x);  sigmoid(2x) = 1/(1+exp(-2x))
    float e = __expf(-2.0f * x);               // v_exp_f32 (transcendental unit)
    float g = __builtin_amdgcn_rcpf(1.0f + e); // v_rcp_f32
    return __builtin_fmaf(x, g, x);            // single v_fma_f32
}

__device__ __forceinline__ v4f gate_vec(v4f a) {
    v4f r;
    r.x = gate_elem(a.x);
    r.y = gate_elem(a.y);
    r.z = gate_elem(a.z);
    r.w = gate_elem(a.w);
    return r;
}

// Exact-cover kernel: grid * 1024 float4s == n4 guaranteed by the host side.
// 256 threads (8 wave32s); each thread does 4 x b128 at +0/+4096/+8192/+12288B
// from one base address -> saddr-form loads with immediate offsets, zero guards.
__global__ __launch_bounds__(256) void TH_gate_v4_exact(const v4f* __restrict__ in,
                                                        v4f* __restrict__ out) {
    const unsigned idx = blockIdx.x * 1024u + threadIdx.x;   // in float4 units

    // Issue all 4 NT loads back-to-back (4 outstanding LOADcnt per wave).
    v4f a0 = __builtin_nontemporal_load(in + idx);
    v4f a1 = __builtin_nontemporal_load(in + idx + 256u);
    v4f a2 = __builtin_nontemporal_load(in + idx + 512u);
    v4f a3 = __builtin_nontemporal_load(in + idx + 768u);

    __builtin_nontemporal_store(gate_vec(a0), out + idx);
    __builtin_nontemporal_store(gate_vec(a1), out + idx + 256u);
    __builtin_nontemporal_store(gate_vec(a2), out + idx + 512u);
    __builtin_nontemporal_store(gate_vec(a3), out + idx + 768u);
}

// Guarded scalar tail for any elements not covered by the exact kernel
// (never used for the fixed 2^25 size, kept for robustness).
__global__ __launch_bounds__(256) void TH_gate_tail(const float* __restrict__ in,
                                                    float* __restrict__ out,
                                                    long long start, long long n) {
    long long i = start + (long long)blockIdx.x * blockDim.x + threadIdx.x;
    if (i < n) {
        out[i] = gate_elem(in[i]);
    }
}

extern "C" void kernel_launch(void* const* d_in, const int* in_sizes, int n_in,
                              void* d_out, int out_size, void* d_ws, size_t ws_size,
                              hipStream_t stream) {
    (void)in_sizes; (void)n_in; (void)d_ws; (void)ws_size;
    const float* x = (const float*)d_in[0];   // only live input; all params are dead code
    float* out = (float*)d_out;

    const long long n  = (long long)out_size;     // 33,554,432
    const long long n4 = n >> 2;                  // 8,388,608 float4s
    const long long per_block = 1024;             // float4s per block (256 thr * 4)

    const long long grid = n4 / per_block;        // 8192 full blocks (exact cover)
    if (grid > 0) {
        TH_gate_v4_exact<<<(unsigned)grid, 256, 0, stream>>>(
            (const v4f*)x, (v4f*)out);
    }

    const long long covered = (grid * per_block) << 2;  // elements handled above
    const long long tail = n - covered;                  // 0 for this workload
    if (tail > 0) {
        const unsigned tgrid = (unsigned)((tail + 255) / 256);
        TH_gate_tail<<<tgrid, 256, 0, stream>>>(x, out, covered, n);
    }
}